// InstanceAttention_42640435315259
// MI455X (gfx1250) — compile-verified
//
#include <hip/hip_runtime.h>

// Problem constants (fixed by reference setup_inputs)
#define D_   256
#define H_   8
#define L_   4
#define HD_  32
#define B_   4
#define L1_  1024
#define L2_  21760   // 128*128 + 64*64 + 32*32 + 16*16

typedef __attribute__((ext_vector_type(2))) float v2f;
typedef __attribute__((ext_vector_type(8))) float v8f;

// ---------------------------------------------------------------------------
// f32 WMMA GEMM with async-LDS A staging:
//   C[M,N] = A[M,256] * W[N,256]^T + bias[N]   (optional per-row zero mask)
//
// K is fixed at 256, so a 16-row A tile is a contiguous 16KB global region.
// Block = 128 threads (4 waves):
//   - all threads stage the 16x256 A tile into LDS with
//     GLOBAL_LOAD_ASYNC_TO_LDS_B128 (ASYNCcnt), then s_wait_asynccnt + barrier
//   - wave w computes the 16(M) x 64(N) slab at n0 = w*64 with 4 accumulators:
//     per K=4 step: 1 ds_load_b64 (A) + 4 global_load_b64 (B) + 4 WMMAs.
// A fragment (16x4 f32): lanes 0-15 -> M=lane,K+0/1; lanes 16-31 -> M=lane-16,K+2/3.
// B fragment (4x16 f32): same K mapping, N = lane&15.
// C/D: VGPR r -> M = m0 + r + 8*(lane>=16), N = n0 + (lane&15).
// ---------------------------------------------------------------------------
__global__ __launch_bounds__(128) void wmma_gemm_nt(
    const float* __restrict__ A, const float* __restrict__ W,
    const float* __restrict__ bias, float* __restrict__ C,
    int M, int N, const unsigned char* __restrict__ mask)
{
    __shared__ __align__(16) float atile[16 * D_];   // 16 KB

    int t    = threadIdx.x;
    int lane = t & 31;
    int wave = t >> 5;
    int m0   = blockIdx.x * 16;

    // ---- async-stage the contiguous 16x256 f32 A tile into LDS ----
    {
        const char* gsrc = (const char*)(A + (size_t)m0 * D_);
        unsigned lbase = (unsigned)(uintptr_t)(&atile[0]);  // low 32 bits = LDS offset
#pragma unroll
        for (int i = 0; i < 8; ++i) {
            unsigned off = (unsigned)t * 16u + (unsigned)i * 2048u;
            unsigned long long g = (unsigned long long)(uintptr_t)(gsrc + off);
            unsigned l = lbase + off;
            asm volatile("global_load_async_to_lds_b128 %0, %1, off"
                         :: "v"(l), "v"(g) : "memory");
        }
        asm volatile("s_wait_asynccnt 0" ::: "memory");
    }
    __syncthreads();

    int n0 = wave * 64;
    if (n0 < N) {                       // wave-uniform: EXEC all-1s inside
        int half = lane >> 4;           // 0: K+0/1,  1: K+2/3
        int lm   = lane & 15;

        const float* arow = atile + lm * D_ + half * 2;            // LDS
        const float* b0 = W + (size_t)(n0 +  0 + lm) * D_ + half * 2;
        const float* b1 = W + (size_t)(n0 + 16 + lm) * D_ + half * 2;
        const float* b2 = W + (size_t)(n0 + 32 + lm) * D_ + half * 2;
        const float* b3 = W + (size_t)(n0 + 48 + lm) * D_ + half * 2;

        v8f c0 = {}, c1 = {}, c2 = {}, c3 = {};
#pragma unroll 2
        for (int k = 0; k < D_; k += 4) {
            v2f a  = *(const v2f*)(arow + k);
            v2f q0 = *(const v2f*)(b0 + k);
            v2f q1 = *(const v2f*)(b1 + k);
            v2f q2 = *(const v2f*)(b2 + k);
            v2f q3 = *(const v2f*)(b3 + k);
            c0 = __builtin_amdgcn_wmma_f32_16x16x4_f32(false, a, false, q0, (short)0, c0, false, false);
            c1 = __builtin_amdgcn_wmma_f32_16x16x4_f32(false, a, false, q1, (short)0, c1, false, false);
            c2 = __builtin_amdgcn_wmma_f32_16x16x4_f32(false, a, false, q2, (short)0, c2, false, false);
            c3 = __builtin_amdgcn_wmma_f32_16x16x4_f32(false, a, false, q3, (short)0, c3, false, false);
        }

#pragma unroll
        for (int sub = 0; sub < 4; ++sub) {
            const v8f& acc = sub == 0 ? c0 : (sub == 1 ? c1 : (sub == 2 ? c2 : c3));
            int col = n0 + sub * 16 + lm;
            float bn = bias ? bias[col] : 0.0f;
#pragma unroll
            for (int r = 0; r < 8; ++r) {
                int row = m0 + half * 8 + r;
                float v = acc[r] + bn;
                if (mask && mask[row]) v = 0.0f;
                C[(size_t)row * N + col] = v;
            }
        }
    }
}

// ---------------------------------------------------------------------------
// Deformable sampling kernel. One block per (b,q); wave w = head h.
// Phase 1: lanes cover the 64 sample points (2 iterations of 32):
//   compute grid coords, repeated attention logits (written to aw output),
//   sw softmax (over all L*KK=64) and lw softmax (over L per kk) via wave32
//   shuffles. Stash {gx, gy, sw, lw} per point in LDS.
// Phase 2: lanes = 32 channels; gather 4 bilinear taps per point from v_proj
//   (coalesced 128B per tap across the wave) and accumulate out/mout heads.
// ---------------------------------------------------------------------------
__global__ __launch_bounds__(256) void deform_sample(
    const float* __restrict__ refw,     // (B,L1,4)
    const float* __restrict__ off_ws,   // (B*L1, 128)  = (h,l,4)
    const float* __restrict__ awlin,    // (B*L1, 128)  = (h,l,2,2)
    const float* __restrict__ vproj,    // (B, L2, H, HD)
    const long long* __restrict__ vshape,   // (L,2)
    const long long* __restrict__ vstart,   // (L,)
    float* __restrict__ aw_out,         // (B*L1, H, L, 4, 4) = (B*L1, 512)
    float* __restrict__ hbuf)           // [out | mout], each (B*L1, 256)
{
    __shared__ float4 smem[H_ * 64];

    int bq   = blockIdx.x;
    int b    = bq / L1_;
    int lane = threadIdx.x & 31;
    int h    = threadIdx.x >> 5;

    float r0 = refw[bq * 4 + 0];
    float r1 = refw[bq * 4 + 1];
    float r2 = refw[bq * 4 + 2];
    float r3 = refw[bq * 4 + 3];

    float gx[2], gy[2], av[2];
#pragma unroll
    for (int it = 0; it < 2; ++it) {
        int p  = it * 32 + lane;       // point id = l*16 + kk
        int l  = p >> 4;
        int kk = p & 15;
        int i  = kk >> 2;              // kernel row
        int j  = kk & 3;               // kernel col
        const float* op = off_ws + (size_t)bq * 128 + h * 16 + l * 4;
        float o0 = op[0], o1 = op[1], o2 = op[2], o3 = op[3];
        float cx = r0 + o0 * 0.125f * r2;
        float cy = r1 + o1 * 0.125f * r3;
        float sx = r2 + o2 * 0.125f * r2;
        float sy = r3 + o3 * 0.125f * r3;
        float kx = ((float)j - 1.5f) * 0.25f;
        float ky = ((float)i - 1.5f) * 0.25f;
        gx[it] = cx + kx * fmaxf(sx, 0.0f);
        gy[it] = cy + ky * fmaxf(sy, 0.0f);
        // repeated attention logit: base index (i/2, j/2) in the 2x2 grid
        float a = awlin[(size_t)bq * 128 + h * 16 + l * 4 + (i >> 1) * 2 + (j >> 1)];
        av[it] = a;
        aw_out[(size_t)bq * 512 + h * 64 + p] = a;   // third output tensor
    }

    // sw: softmax over all 64 points of this head
    float m = fmaxf(av[0], av[1]);
#pragma unroll
    for (int s = 16; s >= 1; s >>= 1) m = fmaxf(m, __shfl_xor(m, s, 32));
    float e0 = __expf(av[0] - m), e1 = __expf(av[1] - m);
    float ssum = e0 + e1;
#pragma unroll
    for (int s = 16; s >= 1; s >>= 1) ssum += __shfl_xor(ssum, s, 32);
    float sw0 = e0 / ssum, sw1 = e1 / ssum;

    // lw: softmax over the 4 levels at fixed kk.
    // lane holds l = (lane>>4) in av[0] and l = 2+(lane>>4) in av[1]; xor-16
    // pairs up the remaining two levels of the same kk.
    float ml = fmaxf(av[0], av[1]);
    ml = fmaxf(ml, __shfl_xor(ml, 16, 32));
    float f0 = __expf(av[0] - ml), f1 = __expf(av[1] - ml);
    float sl = f0 + f1;
    sl += __shfl_xor(sl, 16, 32);
    float lw0 = f0 / sl, lw1 = f1 / sl;

    smem[h * 64 + lane]      = make_float4(gx[0], gy[0], sw0, lw0);
    smem[h * 64 + 32 + lane] = make_float4(gx[1], gy[1], sw1, lw1);
    __syncthreads();

    // Phase 2: lanes = channels, gather + accumulate
    int c = lane;
    float accO = 0.0f, accM = 0.0f;
    for (int l = 0; l < L_; ++l) {
        int Hl = (int)vshape[2 * l];
        int Wl = (int)vshape[2 * l + 1];
        int st = (int)vstart[l];
        for (int kk = 0; kk < 16; ++kk) {
            float4 pt = smem[h * 64 + l * 16 + kk];   // uniform -> LDS broadcast
            float x = pt.x * (float)Wl - 0.5f;
            float y = pt.y * (float)Hl - 0.5f;
            float x0f = floorf(x), y0f = floorf(y);
            int x0 = (int)x0f, y0 = (int)y0f;
            float fx = x - x0f, fy = y - y0f;
            float w00 = (1.0f - fx) * (1.0f - fy);
            float w10 = fx * (1.0f - fy);
            float w01 = (1.0f - fx) * fy;
            float w11 = fx * fy;

            float samp = 0.0f;
            {
                int xi = x0, yi = y0; float w = w00;
                bool ok = (xi >= 0) & (xi < Wl) & (yi >= 0) & (yi < Hl);
                int xc = xi < 0 ? 0 : (xi >= Wl ? Wl - 1 : xi);
                int yc = yi < 0 ? 0 : (yi >= Hl ? Hl - 1 : yi);
                float val = vproj[(size_t)(b * L2_ + st + yc * Wl + xc) * D_ + h * HD_ + c];
                samp += (ok ? w : 0.0f) * val;
            }
            {
                int xi = x0 + 1, yi = y0; float w = w10;
                bool ok = (xi >= 0) & (xi < Wl) & (yi >= 0) & (yi < Hl);
                int xc = xi < 0 ? 0 : (xi >= Wl ? Wl - 1 : xi);
                int yc = yi < 0 ? 0 : (yi >= Hl ? Hl - 1 : yi);
                float val = vproj[(size_t)(b * L2_ + st + yc * Wl + xc) * D_ + h * HD_ + c];
                samp += (ok ? w : 0.0f) * val;
            }
            {
                int xi = x0, yi = y0 + 1; float w = w01;
                bool ok = (xi >= 0) & (xi < Wl) & (yi >= 0) & (yi < Hl);
                int xc = xi < 0 ? 0 : (xi >= Wl ? Wl - 1 : xi);
                int yc = yi < 0 ? 0 : (yi >= Hl ? Hl - 1 : yi);
                float val = vproj[(size_t)(b * L2_ + st + yc * Wl + xc) * D_ + h * HD_ + c];
                samp += (ok ? w : 0.0f) * val;
            }
            {
                int xi = x0 + 1, yi = y0 + 1; float w = w11;
                bool ok = (xi >= 0) & (xi < Wl) & (yi >= 0) & (yi < Hl);
                int xc = xi < 0 ? 0 : (xi >= Wl ? Wl - 1 : xi);
                int yc = yi < 0 ? 0 : (yi >= Hl ? Hl - 1 : yi);
                float val = vproj[(size_t)(b * L2_ + st + yc * Wl + xc) * D_ + h * HD_ + c];
                samp += (ok ? w : 0.0f) * val;
            }
            accO += samp * pt.z;
            accM += samp * pt.w;
        }
    }
    hbuf[(size_t)bq * D_ + h * HD_ + c] = accO;
    hbuf[(size_t)(B_ * L1_) * D_ + (size_t)bq * D_ + h * HD_ + c] = accM;
}

// ---------------------------------------------------------------------------
extern "C" void kernel_launch(void* const* d_in, const int* in_sizes, int n_in,
                              void* d_out, int out_size, void* d_ws, size_t ws_size,
                              hipStream_t stream) {
    const float* query  = (const float*)d_in[0];   // (B,L1,256)
    const float* value  = (const float*)d_in[1];   // (B,L2,256)
    const float* refw   = (const float*)d_in[2];   // (B,L1,4)
    const float* Wv     = (const float*)d_in[3];   // (256,256)
    const float* bv     = (const float*)d_in[4];
    const float* Wo     = (const float*)d_in[5];   // (256,256)
    const float* bo     = (const float*)d_in[6];
    const float* Wbox   = (const float*)d_in[7];   // (128,256)
    const float* bbox   = (const float*)d_in[8];
    const float* Wattn  = (const float*)d_in[9];   // (128,256)
    const float* battn  = (const float*)d_in[10];
    const long long* vshape = (const long long*)d_in[11];
    const long long* vstart = (const long long*)d_in[12];
    const unsigned char* vmask = (const unsigned char*)d_in[13];   // (B,L2) bool

    float* ws    = (float*)d_ws;
    float* vproj = ws;                                   // B*L2*256 = 22,282,240 f
    float* offw  = vproj + (size_t)B_ * L2_ * D_;        // B*L1*128 =    524,288 f
    float* awlin = offw + (size_t)B_ * L1_ * 128;        // B*L1*128 =    524,288 f
    float* hbuf  = awlin + (size_t)B_ * L1_ * 128;       // 2*B*L1*256 = 2,097,152 f

    float* out = (float*)d_out;                          // [out | mout | aw]
    float* aw_out = out + 2 * (B_ * L1_ * D_);

    dim3 blk(128);

    // 1) value projection (masked):  v_proj = value . Wv^T + bv
    wmma_gemm_nt<<<dim3((B_ * L2_) / 16), blk, 0, stream>>>(
        value, Wv, bv, vproj, B_ * L2_, D_, vmask);

    // 2) box offsets:  off = query . Wbox^T + bbox   (N = 128)
    wmma_gemm_nt<<<dim3((B_ * L1_) / 16), blk, 0, stream>>>(
        query, Wbox, bbox, offw, B_ * L1_, 128, nullptr);

    // 3) attention logits:  awlin = query . Wattn^T + battn   (N = 128)
    wmma_gemm_nt<<<dim3((B_ * L1_) / 16), blk, 0, stream>>>(
        query, Wattn, battn, awlin, B_ * L1_, 128, nullptr);

    // 4) geometry + softmaxes + bilinear gather + head accumulation
    deform_sample<<<dim3(B_ * L1_), dim3(256), 0, stream>>>(
        refw, offw, awlin, vproj, vshape, vstart, aw_out, hbuf);

    // 5) output projection for [out; mout]:  (8192 x 256) . Wo^T + bo -> d_out
    wmma_gemm_nt<<<dim3((2 * B_ * L1_) / 16), blk, 0, stream>>>(
        hbuf, Wo, bo, out, 2 * B_ * L1_, D_, nullptr);
}